// CostVolumeModule_15865609191852
// MI455X (gfx1250) — compile-verified
//
#include <hip/hip_runtime.h>
#include <math.h>

// Problem constants (fixed by setup_inputs)
constexpr int B_ = 2, F_ = 2, C_ = 3, H_ = 256, W_ = 512, D_ = 32;
constexpr int TILE = 16, IT = 20, PT = 18;     // out tile, input tile (halo 2), pool tile (halo 1)
constexpr float EPSF  = 1e-7f;
constexpr float C1_   = 0.0001f;               // 0.01^2
constexpr float C2_   = 0.0009f;               // 0.03^2
constexpr float ALPHA_ = 10.0f;
constexpr int BORDER_ = 2;

typedef __attribute__((ext_vector_type(2))) float v2f;
typedef __attribute__((ext_vector_type(8))) float v8f;

// ---------------- 4x4 inverse (adjugate; layout-agnostic) ----------------
__device__ void inv4(const float* m, float* out) {
    float inv[16];
    inv[0]  =  m[5]*m[10]*m[15] - m[5]*m[11]*m[14] - m[9]*m[6]*m[15] + m[9]*m[7]*m[14] + m[13]*m[6]*m[11] - m[13]*m[7]*m[10];
    inv[4]  = -m[4]*m[10]*m[15] + m[4]*m[11]*m[14] + m[8]*m[6]*m[15] - m[8]*m[7]*m[14] - m[12]*m[6]*m[11] + m[12]*m[7]*m[10];
    inv[8]  =  m[4]*m[9]*m[15]  - m[4]*m[11]*m[13] - m[8]*m[5]*m[15] + m[8]*m[7]*m[13] + m[12]*m[5]*m[11] - m[12]*m[7]*m[9];
    inv[12] = -m[4]*m[9]*m[14]  + m[4]*m[10]*m[13] + m[8]*m[5]*m[14] - m[8]*m[6]*m[13] - m[12]*m[5]*m[10] + m[12]*m[6]*m[9];
    inv[1]  = -m[1]*m[10]*m[15] + m[1]*m[11]*m[14] + m[9]*m[2]*m[15] - m[9]*m[3]*m[14] - m[13]*m[2]*m[11] + m[13]*m[3]*m[10];
    inv[5]  =  m[0]*m[10]*m[15] - m[0]*m[11]*m[14] - m[8]*m[2]*m[15] + m[8]*m[3]*m[14] + m[12]*m[2]*m[11] - m[12]*m[3]*m[10];
    inv[9]  = -m[0]*m[9]*m[15]  + m[0]*m[11]*m[13] + m[8]*m[1]*m[15] - m[8]*m[3]*m[13] - m[12]*m[1]*m[11] + m[12]*m[3]*m[9];
    inv[13] =  m[0]*m[9]*m[14]  - m[0]*m[10]*m[13] - m[8]*m[1]*m[14] + m[8]*m[2]*m[13] + m[12]*m[1]*m[10] - m[12]*m[2]*m[9];
    inv[2]  =  m[1]*m[6]*m[15]  - m[1]*m[7]*m[14]  - m[5]*m[2]*m[15] + m[5]*m[3]*m[14] + m[13]*m[2]*m[7]  - m[13]*m[3]*m[6];
    inv[6]  = -m[0]*m[6]*m[15]  + m[0]*m[7]*m[14]  + m[4]*m[2]*m[15] - m[4]*m[3]*m[14] - m[12]*m[2]*m[7]  + m[12]*m[3]*m[6];
    inv[10] =  m[0]*m[5]*m[15]  - m[0]*m[7]*m[13]  - m[4]*m[1]*m[15] + m[4]*m[3]*m[13] + m[12]*m[1]*m[7]  - m[12]*m[3]*m[5];
    inv[14] = -m[0]*m[5]*m[14]  + m[0]*m[6]*m[13]  + m[4]*m[1]*m[14] - m[4]*m[2]*m[13] - m[12]*m[1]*m[6]  + m[12]*m[2]*m[5];
    inv[3]  = -m[1]*m[6]*m[11]  + m[1]*m[7]*m[10]  + m[5]*m[2]*m[11] - m[5]*m[3]*m[10] - m[9]*m[2]*m[7]   + m[9]*m[3]*m[6];
    inv[7]  =  m[0]*m[6]*m[11]  - m[0]*m[7]*m[10]  - m[4]*m[2]*m[11] + m[4]*m[3]*m[10] + m[8]*m[2]*m[7]   - m[8]*m[3]*m[6];
    inv[11] = -m[0]*m[5]*m[11]  + m[0]*m[7]*m[9]   + m[4]*m[1]*m[11] - m[4]*m[3]*m[9]  - m[8]*m[1]*m[7]   + m[8]*m[3]*m[5];
    inv[15] =  m[0]*m[5]*m[10]  - m[0]*m[6]*m[9]   - m[4]*m[1]*m[10] + m[4]*m[2]*m[9]  + m[8]*m[1]*m[6]   - m[8]*m[2]*m[5];
    float det = m[0]*inv[0] + m[1]*inv[4] + m[2]*inv[8] + m[3]*inv[12];
    float id = 1.0f / det;
    for (int i = 0; i < 16; ++i) out[i] = inv[i] * id;
}

// ws layout (floats): [0, B*9) invK3 per batch; [B*9, B*9 + F*B*12) P (3x4) per (f,b)
__global__ void cv_setup(const float* __restrict__ kfK, const float* __restrict__ kfPose,
                         const float* __restrict__ fK,  const float* __restrict__ fPose,
                         float* __restrict__ ws) {
    int t = threadIdx.x;
    if (t < B_) {
        float inv[16]; inv4(kfK + t * 16, inv);
        for (int i = 0; i < 3; ++i)
            for (int j = 0; j < 3; ++j)
                ws[t * 9 + i * 3 + j] = inv[i * 4 + j];
    }
    if (t < F_ * B_) {
        int f = t / B_, b = t % B_;
        float ip[16]; inv4(fPose + (f * B_ + b) * 16, ip);
        const float* kp = kfPose + b * 16;
        float T[16];
        for (int i = 0; i < 4; ++i)
            for (int j = 0; j < 4; ++j) {
                float s = 0.f;
                for (int k = 0; k < 4; ++k) s += ip[i * 4 + k] * kp[k * 4 + j];
                T[i * 4 + j] = s;
            }
        const float* K = fK + (f * B_ + b) * 16;
        float* P = ws + B_ * 9 + (f * B_ + b) * 12;
        for (int i = 0; i < 3; ++i)
            for (int j = 0; j < 4; ++j) {
                float s = 0.f;
                for (int k = 0; k < 4; ++k) s += K[i * 4 + k] * T[k * 4 + j];
                P[i * 4 + j] = s;
            }
    }
}

__device__ __forceinline__ int reflect_idx(int i, int n) {
    if (i < 0) i = -i;
    if (i > n - 1) i = 2 * (n - 1) - i;
    return i;
}

// ------------- fused warp + SSIM + box-sum (WMMA) per 16x16 tile -------------
// grid: (W/16, H/16, B*F*D), block: 32 (one wave, EXEC all ones at the WMMA)
__launch_bounds__(32)
__global__ void cv_sad_kernel(const float* __restrict__ keyframe,
                              const float* __restrict__ frames,
                              const float* __restrict__ depths,
                              const float* __restrict__ ws,
                              float* __restrict__ sadOut /* [F,B,D,H,W] */) {
    __shared__ float ktile[C_][IT][IT];
    __shared__ float wtile[C_][IT][IT];
    __shared__ float wsum[PT][PT];
    __shared__ float hsum[20][TILE];   // rows 18,19 are zero K-padding for WMMA

    const int lane = threadIdx.x;
    const int w0 = blockIdx.x * TILE;
    const int h0 = blockIdx.y * TILE;
    const int plane = blockIdx.z;
    const int d = plane % D_;
    const int f = (plane / D_) % F_;
    const int b = plane / (D_ * F_);

    float ik[9], P[12];
    for (int i = 0; i < 9; ++i)  ik[i] = ws[b * 9 + i];
    for (int i = 0; i < 12; ++i) P[i]  = ws[B_ * 9 + (f * B_ + b) * 12 + i];

    // keyframe tile, reflect-padded at image borders, +0.5 (SSIM input)
    for (int i = lane; i < C_ * IT * IT; i += 32) {
        int c = i / (IT * IT), idx = i % (IT * IT);
        int tr = idx / IT, tc = idx % IT;
        int r  = reflect_idx(h0 + tr - 2, H_);
        int cc = reflect_idx(w0 + tc - 2, W_);
        ktile[c][tr][tc] = keyframe[((size_t)(b * C_ + c) * H_ + r) * W_ + cc] + 0.5f;
    }

    // warped tile: project reflected pixel coords at this depth, bilinear (zero pad), +0.5
    const float* fimg = frames + (size_t)(f * B_ + b) * C_ * H_ * W_;
    const float* dep  = depths + (size_t)(b * D_ + d) * H_ * W_;
    for (int i = lane; i < IT * IT; i += 32) {
        int tr = i / IT, tc = i % IT;
        int r  = reflect_idx(h0 + tr - 2, H_);
        int cc = reflect_idx(w0 + tc - 2, W_);
        float depth = dep[r * W_ + cc];
        float x = (float)cc, y = (float)r;
        float rx = ik[0] * x + ik[1] * y + ik[2];
        float ry = ik[3] * x + ik[4] * y + ik[5];
        float rz = ik[6] * x + ik[7] * y + ik[8];
        float qx = depth * rx, qy = depth * ry, qz = depth * rz;
        float cpx = P[0] * qx + P[1] * qy + P[2]  * qz + P[3];
        float cpy = P[4] * qx + P[5] * qy + P[6]  * qz + P[7];
        float cpz = P[8] * qx + P[9] * qy + P[10] * qz + P[11];
        float iz = 1.0f / (cpz + EPSF);
        float px = cpx * iz, py = cpy * iz;
        float gx = fminf(fmaxf((px / (float)(W_ - 1) - 0.5f) * 2.0f, -2.0f), 2.0f);
        float gy = fminf(fmaxf((py / (float)(H_ - 1) - 0.5f) * 2.0f, -2.0f), 2.0f);
        float xs  = (gx + 1.0f) * (W_ * 0.5f) - 0.5f;
        float ysf = (gy + 1.0f) * (H_ * 0.5f) - 0.5f;
        float x0f = floorf(xs), y0f = floorf(ysf);
        float wx = xs - x0f, wy = ysf - y0f;
        int x0 = (int)x0f, y0 = (int)y0f;
        float w00 = (1.f - wx) * (1.f - wy), w10 = wx * (1.f - wy);
        float w01 = (1.f - wx) * wy,         w11 = wx * wy;
        bool v00 = (x0     >= 0) && (x0     <= W_ - 1) && (y0     >= 0) && (y0     <= H_ - 1);
        bool v10 = (x0 + 1 >= 0) && (x0 + 1 <= W_ - 1) && (y0     >= 0) && (y0     <= H_ - 1);
        bool v01 = (x0     >= 0) && (x0     <= W_ - 1) && (y0 + 1 >= 0) && (y0 + 1 <= H_ - 1);
        bool v11 = (x0 + 1 >= 0) && (x0 + 1 <= W_ - 1) && (y0 + 1 >= 0) && (y0 + 1 <= H_ - 1);
        int xc0 = min(max(x0, 0), W_ - 1),     xc1 = min(max(x0 + 1, 0), W_ - 1);
        int yc0 = min(max(y0, 0), H_ - 1),     yc1 = min(max(y0 + 1, 0), H_ - 1);
        for (int c = 0; c < C_; ++c) {
            const float* img = fimg + (size_t)c * H_ * W_;
            float t00 = v00 ? img[(size_t)yc0 * W_ + xc0] : 0.0f;
            float t10 = v10 ? img[(size_t)yc0 * W_ + xc1] : 0.0f;
            float t01 = v01 ? img[(size_t)yc1 * W_ + xc0] : 0.0f;
            float t11 = v11 ? img[(size_t)yc1 * W_ + xc1] : 0.0f;
            wtile[c][tr][tc] = t00 * w00 + t10 * w10 + t01 * w01 + t11 * w11 + 0.5f;
        }
    }
    __syncthreads();

    // SSIM (3x3 reflect-mean stats in one 9-tap pass) + channel-weighted sum, at 18x18
    const float cw[3] = {5.0f / 288.0f, 16.0f / 288.0f, 11.0f / 288.0f}; // CHANNEL_W/9
    for (int i = lane; i < PT * PT; i += 32) {
        int tr = i / PT, tc = i % PT;
        int r = h0 + tr - 1, cc = w0 + tc - 1;
        float outv = 0.0f;
        if (r >= 0 && r < H_ && cc >= 0 && cc < W_) {  // zero-pad for the SAD box sum
            for (int c = 0; c < C_; ++c) {
                float sx = 0.f, sy = 0.f, sxx = 0.f, syy = 0.f, sxy = 0.f;
                #pragma unroll
                for (int dy = 0; dy < 3; ++dy)
                    #pragma unroll
                    for (int dx = 0; dx < 3; ++dx) {
                        float xv = wtile[c][tr + dy][tc + dx];
                        float yv = ktile[c][tr + dy][tc + dx];
                        sx += xv; sy += yv;
                        sxx += xv * xv; syy += yv * yv; sxy += xv * yv;
                    }
                const float inv9 = 1.0f / 9.0f;
                float mux = sx * inv9, muy = sy * inv9;
                float sigx  = sxx * inv9 - mux * mux;
                float sigy  = syy * inv9 - muy * muy;
                float sigxy = sxy * inv9 - mux * muy;
                float n   = (2.0f * mux * muy + C1_) * (2.0f * sigxy + C2_);
                float den = (mux * mux + muy * muy + C1_) * (sigx + sigy + C2_);
                float diff = fminf(fmaxf((1.0f - n / den) * 0.5f, 0.0f), 1.0f);
                outv += cw[c] * diff;
            }
        }
        wsum[tr][tc] = outv;
    }
    __syncthreads();

    // horizontal 3-tap sums; rows 18..19 zeroed (K padding)
    for (int i = lane; i < 20 * TILE; i += 32) {
        int tr = i / TILE, j = i % TILE;
        hsum[tr][j] = (tr < PT) ? (wsum[tr][j] + wsum[tr][j + 1] + wsum[tr][j + 2]) : 0.0f;
    }
    __syncthreads();

    // vertical 3-tap sums on the matrix pipe:
    // sad(16x16) = A(16x20) * hsum(20x16), A[m][k] = (m <= k <= m+2)
    // 5 chained V_WMMA_F32_16X16X4_F32 (K = 4 each)
    const int half = lane >> 4;    // 0: lanes 0-15, 1: lanes 16-31
    const int lm   = lane & 15;
    v8f acc = {};
    #pragma unroll
    for (int k0 = 0; k0 < 20; k0 += 4) {
        int ka = k0 + half * 2;    // ISA A-layout: VGPR0 holds K=k0 / k0+2, VGPR1 K=k0+1 / k0+3
        v2f a, bb;
        a.x = (ka     >= lm && ka     <= lm + 2) ? 1.0f : 0.0f;
        a.y = (ka + 1 >= lm && ka + 1 <= lm + 2) ? 1.0f : 0.0f;
        bb.x = hsum[ka][lm];
        bb.y = hsum[ka + 1][lm];
        acc = __builtin_amdgcn_wmma_f32_16x16x4_f32(
            /*neg_a=*/false, a, /*neg_b=*/false, bb,
            /*c_mod=*/(short)0, acc, /*reuse_a=*/false, /*reuse_b=*/false);
    }

    // D layout: VGPR v -> M = v + 8*half, N = lm
    float* sadp = sadOut + (size_t)((f * B_ + b) * D_ + d) * H_ * W_;
    #pragma unroll
    for (int v = 0; v < 8; ++v) {
        int m = v + half * 8;
        sadp[(size_t)(h0 + m) * W_ + (w0 + lm)] = acc[v];
    }
}

// ------------- per-pixel mask/weight reduction, cv + sfcv outputs -------------
__launch_bounds__(256)
__global__ void cv_finalize(const float* __restrict__ depths,
                            const float* __restrict__ ws,
                            float* __restrict__ cvOut   /* [B,D,H,W] */,
                            float* __restrict__ sfcvOut /* [F,B,D,H,W]: holds sad on input */) {
    int p = blockIdx.x * blockDim.x + threadIdx.x;
    if (p >= H_ * W_) return;
    int b = blockIdx.y;
    int h = p / W_, w = p % W_;

    float ik[9];
    for (int i = 0; i < 9; ++i) ik[i] = ws[b * 9 + i];
    float x = (float)w, y = (float)h;
    float rx = ik[0] * x + ik[1] * y + ik[2];
    float ry = ik[3] * x + ik[4] * y + ik[5];
    float rz = ik[6] * x + ik[7] * y + ik[8];

    float selfMask = (h >= BORDER_ && h < H_ - BORDER_ && w >= BORDER_ && w < W_ - BORDER_) ? 1.0f : 0.0f;

    float sad[F_][D_];
    float wmsk[F_];
    for (int f = 0; f < F_; ++f) {
        float P[12];
        for (int i = 0; i < 12; ++i) P[i] = ws[B_ * 9 + (f * B_ + b) * 12 + i];
        float wm = selfMask;
        for (int d = 0; d < D_; ++d) {
            float depth = depths[((size_t)(b * D_ + d) * H_ + h) * W_ + w];
            float qx = depth * rx, qy = depth * ry, qz = depth * rz;
            float cpx = P[0] * qx + P[1] * qy + P[2]  * qz + P[3];
            float cpy = P[4] * qx + P[5] * qy + P[6]  * qz + P[7];
            float cpz = P[8] * qx + P[9] * qy + P[10] * qz + P[11];
            float iz = 1.0f / (cpz + EPSF);
            float px = cpx * iz, py = cpy * iz;
            float gx = fminf(fmaxf((px / (float)(W_ - 1) - 0.5f) * 2.0f, -2.0f), 2.0f);
            float gy = fminf(fmaxf((py / (float)(H_ - 1) - 0.5f) * 2.0f, -2.0f), 2.0f);
            float xs  = (gx + 1.0f) * (W_ * 0.5f) - 0.5f;
            float ysf = (gy + 1.0f) * (H_ * 0.5f) - 0.5f;
            float x0f = floorf(xs), y0f = floorf(ysf);
            float wx = xs - x0f, wy = ysf - y0f;
            int x0 = (int)x0f, y0 = (int)y0f;
            auto mval = [&](int xi, int yi) -> float {
                bool vimg = (xi >= 0) && (xi <= W_ - 1) && (yi >= 0) && (yi <= H_ - 1);
                bool vbox = (xi >= BORDER_) && (xi < W_ - BORDER_) && (yi >= BORDER_) && (yi < H_ - BORDER_);
                return (vimg && vbox) ? 1.0f : 0.0f;
            };
            float v = mval(x0,     y0)     * (1.f - wx) * (1.f - wy)
                    + mval(x0 + 1, y0)     * wx         * (1.f - wy)
                    + mval(x0,     y0 + 1) * (1.f - wx) * wy
                    + mval(x0 + 1, y0 + 1) * wx         * wy;
            wm *= (v != 0.0f) ? 1.0f : 0.0f;
        }
        wmsk[f] = wm;
        const float* sp = sfcvOut + ((size_t)((f * B_ + b) * D_) * H_ + h) * W_ + w;
        for (int d = 0; d < D_; ++d)
            sad[f][d] = sp[(size_t)d * H_ * W_];
    }

    float weight[F_];
    for (int f = 0; f < F_; ++f) {
        float mn = sad[f][0];
        for (int d = 1; d < D_; ++d) mn = fminf(mn, sad[f][d]);
        float ssum = 0.0f;
        for (int d = 0; d < D_; ++d) {
            float t = sad[f][d] - mn;
            ssum += expf(-ALPHA_ * t * t);
        }
        weight[f] = (1.0f - (ssum - 1.0f) / (float)(D_ - 1)) * wmsk[f];
    }
    float wsumf = 0.0f;
    for (int f = 0; f < F_; ++f) wsumf += weight[f];
    bool zero = (wsumf == 0.0f);
    float invw = zero ? 0.0f : 1.0f / wsumf;

    for (int d = 0; d < D_; ++d) {
        float acc = 0.0f;
        for (int f = 0; f < F_; ++f) acc += sad[f][d] * weight[f];
        float cvv = zero ? 0.0f : (1.0f - 2.0f * acc * invw);
        cvOut[((size_t)(b * D_ + d) * H_ + h) * W_ + w] = cvv;
    }
    for (int f = 0; f < F_; ++f) {
        float* sp = sfcvOut + ((size_t)((f * B_ + b) * D_) * H_ + h) * W_ + w;
        for (int d = 0; d < D_; ++d)
            sp[(size_t)d * H_ * W_] = (1.0f - 2.0f * sad[f][d]) * wmsk[f];
    }
}

extern "C" void kernel_launch(void* const* d_in, const int* in_sizes, int n_in,
                              void* d_out, int out_size, void* d_ws, size_t ws_size,
                              hipStream_t stream) {
    const float* keyframe = (const float*)d_in[0];  // [B,C,H,W]
    const float* kfK      = (const float*)d_in[1];  // [B,4,4]
    const float* kfPose   = (const float*)d_in[2];  // [B,4,4]
    const float* frames   = (const float*)d_in[3];  // [F,B,C,H,W]
    const float* fK       = (const float*)d_in[4];  // [F,B,4,4]
    const float* fPose    = (const float*)d_in[5];  // [F,B,4,4]
    const float* depths   = (const float*)d_in[6];  // [B,D,H,W]

    float* cvOut   = (float*)d_out;                            // [B,D,H,W]
    float* sfcvOut = cvOut + (size_t)B_ * D_ * H_ * W_;        // [F,B,D,H,W]
    float* wsf     = (float*)d_ws;                             // 66 floats of matrices

    cv_setup<<<1, 32, 0, stream>>>(kfK, kfPose, fK, fPose, wsf);

    dim3 gb(W_ / TILE, H_ / TILE, B_ * F_ * D_);
    cv_sad_kernel<<<gb, 32, 0, stream>>>(keyframe, frames, depths, wsf, sfcvOut);

    dim3 gc((H_ * W_ + 255) / 256, B_);
    cv_finalize<<<gc, 256, 0, stream>>>(depths, wsf, cvOut, sfcvOut);
}